// AetherLoss_69045894250969
// MI455X (gfx1250) — compile-verified
//
#include <hip/hip_runtime.h>
#include <hip/hip_bf16.h>

typedef __attribute__((ext_vector_type(2))) float v2f;
typedef __attribute__((ext_vector_type(8))) float v8f;

#define NPTS     4096
#define NBATCH   8
#define NLAT     2048                  // 8 * 256 latent elements
#define RPW      32                    // rows per wave (2 row fragments)
#define ROWCHUNK 256                   // rows per WG (8 waves x 32)
#define COLCHUNK 512                   // cols per WG
#define ROWCS    (NPTS / ROWCHUNK)     // 16
#define COLCS    (NPTS / COLCHUNK)     // 8
#define NTILES   (COLCHUNK / 16)       // 32 tile iterations per wave
#define CW       (COLCHUNK + 16)       // padded colw slice stride (bank de-conflict)

// Monotone bijection float -> u32 so unsigned atomicMin == float min.
__device__ __forceinline__ unsigned fflip(float f) {
  unsigned u = __float_as_uint(f);
  return (u & 0x80000000u) ? ~u : (u | 0x80000000u);
}
__device__ __forceinline__ float funflip(unsigned u) {
  return __uint_as_float((u & 0x80000000u) ? (u ^ 0x80000000u) : ~u);
}

// Bare CDNA5 min without clang's canonicalize-of-inputs (safe: data is finite).
__device__ __forceinline__ void vmin_acc(float& acc, float v) {
  asm("v_min_num_f32 %0, %0, %1" : "+v"(acc) : "v"(v));
}

__global__ __launch_bounds__(256) void aether_init_kernel(unsigned* __restrict__ mins_u) {
  int i = blockIdx.x * 256 + threadIdx.x;
  if (i < 2 * NBATCH * NPTS) mins_u[i] = 0xFFFFFFFFu;
}

// A = recon_x (xp), Bp = x (yp), both [B,3,N] channels-first f32.
__global__ __launch_bounds__(256) void aether_chamfer_kernel(
    const float* __restrict__ A, const float* __restrict__ Bp,
    unsigned* __restrict__ minx_u, unsigned* __restrict__ miny_u) {
  __shared__ float2 bf1[(COLCHUNK + 16) * 2];  // pre-swizzled B fragments: {y0,y1} | {y2,|y|^2}
  __shared__ float  y2l[COLCHUNK];             // |y|^2 for the flush correction
  __shared__ float  colw[16 * CW];             // per (wave,half) col-min slices: plain stores

  const int tid = threadIdx.x;
  const int blk = blockIdx.x;
  const int b    = blk >> 7;                   // 128 WGs per batch
  const int rem  = blk & 127;
  const int rowc = rem >> 3;                   // 16 row chunks
  const int colc = rem & 7;                    // 8 col chunks
  const int m_base = colc * COLCHUNK;
  const float* xb = A  + b * 3 * NPTS;
  const float* yb = Bp + b * 3 * NPTS;

  // ---- stage this WG's y column chunk, pre-swizzled per K-half ----
  for (int i = tid; i < COLCHUNK; i += 256) {
    const int m = m_base + i;
    float y0 = yb[m], y1 = yb[NPTS + m], y2c = yb[2 * NPTS + m];
    float y2s = y0 * y0 + y1 * y1 + y2c * y2c;
    bf1[2 * i]     = make_float2(y0, y1);      // lanes 0-15: K0,K1
    bf1[2 * i + 1] = make_float2(y2c, y2s);    // lanes 16-31: K2,K3=|y|^2
    y2l[i] = y2s;
  }
  __syncthreads();

  const int wave = tid >> 5;
  const int lane = tid & 31;
  const int half = lane >> 4;                  // wave32: lane halves carry K {0,1} / {2,3}
  const int row  = lane & 15;
  const int n0   = rowc * ROWCHUNK + wave * RPW;
  const int nA   = n0 + row;
  const int nB   = n0 + 16 + row;
  const int cwbase = ((wave << 1) + half) * CW;

  // Raw coords for the two 16-row fragments.
  float xA0 = half ? xb[2 * NPTS + nA] : xb[nA];
  float xA1 = half ? 0.0f : xb[NPTS + nA];
  float xB0 = half ? xb[2 * NPTS + nB] : xb[nB];
  float xB1 = half ? 0.0f : xb[NPTS + nB];

  // Full |x_n|^2 per lane (combine the two K-halves).
  float pA = xA0 * xA0 + xA1 * xA1;
  float pB = xB0 * xB0 + xB1 * xB1;
  float x2fA = pA + __shfl_xor(pA, 16, 32);
  float x2fB = pB + __shfl_xor(pB, 16, 32);

  // A fragments with rank-1 terms folded into the free K=3 slot:
  //  A1 = {-2x0,-2x1,-2x2, 1}      vs B1={y,|y|^2} -> |y|^2 - 2<x,y>  (row operand)
  //  A2 = {-2x0,-2x1,-2x2, |x|^2}  vs B2={y, 1}    -> |x|^2 - 2<x,y>  (col operand)
  v2f avA1 = {-2.0f * xA0, half ? 1.0f : -2.0f * xA1};
  v2f avA2 = {-2.0f * xA0, half ? x2fA : -2.0f * xA1};
  v2f avB1 = {-2.0f * xB0, half ? 1.0f : -2.0f * xB1};
  v2f avB2 = {-2.0f * xB0, half ? x2fB : -2.0f * xB1};

  // x2v[g] = |x|^2 of row (g + 8*half): only for the final row-min correction.
  float x2vA[8], x2vB[8];
#pragma unroll
  for (int g = 0; g < 8; ++g) {
    x2vA[g] = __shfl(x2fA, (half << 3) + g, 32);
    x2vB[g] = __shfl(x2fB, (half << 3) + g, 32);
  }

  float raccA[8], raccB[8];
#pragma unroll
  for (int g = 0; g < 8; ++g) { raccA[g] = __builtin_inff(); raccB[g] = __builtin_inff(); }

  // ---- main loop: 1 ds_load_b64 feeds 4 WMMAs; epilogue is pure mins ----
  float2 fb = bf1[(row << 1) + half];
#pragma unroll 2
  for (int t = 0; t < NTILES; ++t) {
    float2 fn = bf1[((((t + 1) << 4) + row) << 1) + half];  // prefetch (pad covers last)

    v2f bv1 = {fb.x, fb.y};                    // direct from LDS: no selects
    v2f bv2 = {fb.x, half ? 1.0f : fb.y};      // one cndmask

    v8f c1A = {}, c1B = {}, c2A = {}, c2B = {};
    c1A = __builtin_amdgcn_wmma_f32_16x16x4_f32(false, avA1, false, bv1, (short)0, c1A, false, false);
    c1B = __builtin_amdgcn_wmma_f32_16x16x4_f32(false, avB1, false, bv1, (short)0, c1B, false, false);
    c2A = __builtin_amdgcn_wmma_f32_16x16x4_f32(false, avA2, false, bv2, (short)0, c2A, false, false);
    c2B = __builtin_amdgcn_wmma_f32_16x16x4_f32(false, avB2, false, bv2, (short)0, c2B, false, false);

    // Row path: c1 already equals |y|^2 - 2<x,y>; bare v_min_num, no canonicalize.
#pragma unroll
    for (int g = 0; g < 8; ++g) {
      vmin_acc(raccA[g], c1A[g]);
      vmin_acc(raccB[g], c1B[g]);
    }
    // Col path: c2 already equals |x|^2 - 2<x,y>; per-half partial, min3 chain.
    float cmin = fminf(c2A[0], c2A[1]);
#pragma unroll
    for (int g = 2; g < 8; g += 2) cmin = fminf(cmin, fminf(c2A[g], c2A[g + 1]));
#pragma unroll
    for (int g = 0; g < 8; g += 2) cmin = fminf(cmin, fminf(c2B[g], c2B[g + 1]));

    colw[cwbase + (t << 4) + row] = cmin;      // all 32 lanes, distinct slots, full EXEC
    fb = fn;
  }

  // ---- row mins over this col chunk -> global atomic min across the 8 chunks ----
#pragma unroll
  for (int g = 0; g < 8; ++g) {
    float rA = raccA[g], rB = raccB[g];
    rA = fminf(rA, __shfl_xor(rA, 8, 32));
    rB = fminf(rB, __shfl_xor(rB, 8, 32));
    rA = fminf(rA, __shfl_xor(rA, 4, 32));
    rB = fminf(rB, __shfl_xor(rB, 4, 32));
    rA = fminf(rA, __shfl_xor(rA, 2, 32));
    rB = fminf(rB, __shfl_xor(rB, 2, 32));
    rA = fminf(rA, __shfl_xor(rA, 1, 32));
    rB = fminf(rB, __shfl_xor(rB, 1, 32));
    raccA[g] = rA + x2vA[g];
    raccB[g] = rB + x2vB[g];
  }
  if (row == 0) {   // lanes 0 and 16 carry the two M-halves of each fragment
#pragma unroll
    for (int g = 0; g < 8; ++g) {
      atomicMin(&minx_u[b * NPTS + n0 + (half << 3) + g],      fflip(raccA[g]));
      atomicMin(&minx_u[b * NPTS + n0 + 16 + (half << 3) + g], fflip(raccB[g]));
    }
  }

  // ---- flush: 16-way (wave x half) col min (+|y|^2) -> global atomic min ----
  __syncthreads();
  for (int i = tid; i < COLCHUNK; i += 256) {
    float v = colw[i];
#pragma unroll
    for (int w = 1; w < 16; ++w) v = fminf(v, colw[w * CW + i]);
    atomicMin(&miny_u[b * NPTS + m_base + i], fflip(v + y2l[i]));
  }
}

__global__ __launch_bounds__(256) void aether_finalize_kernel(
    const float* __restrict__ mu, const float* __restrict__ logvar,
    const unsigned* __restrict__ minx_u, const unsigned* __restrict__ miny_u,
    float* __restrict__ out) {
  __shared__ float red[256];
  const int tid = threadIdx.x;

  float acc = 0.0f;   // sum of min_x over all rows + sum of min_y over all cols
  for (int i = tid; i < NBATCH * NPTS; i += 256)
    acc += funflip(minx_u[i]) + funflip(miny_u[i]);

  float kacc = 0.0f;  // sum(1 + logvar - mu^2 - exp(logvar))
  for (int i = tid; i < NLAT; i += 256) {
    float m_ = mu[i], lv = logvar[i];
    kacc += 1.0f + lv - m_ * m_ - expf(lv);
  }

  red[tid] = acc;
  __syncthreads();
  for (int s2 = 128; s2 > 0; s2 >>= 1) {
    if (tid < s2) red[tid] += red[tid + s2];
    __syncthreads();
  }
  float chamfer_sum = red[0];
  __syncthreads();
  red[tid] = kacc;
  __syncthreads();
  for (int s2 = 128; s2 > 0; s2 >>= 1) {
    if (tid < s2) red[tid] += red[tid + s2];
    __syncthreads();
  }
  if (tid == 0) {
    float ksum  = red[0];
    float kld   = -0.5f * ksum / (float)NBATCH;
    float recon = chamfer_sum * (1.0f / (float)(NBATCH * NPTS));
    out[0] = recon + 1.0f * kld;   // total (BETA = 1.0)
    out[1] = recon;
    out[2] = kld;
  }
}

extern "C" void kernel_launch(void* const* d_in, const int* in_sizes, int n_in,
                              void* d_out, int out_size, void* d_ws, size_t ws_size,
                              hipStream_t stream) {
  (void)in_sizes; (void)n_in; (void)out_size; (void)ws_size;
  const float* recon  = (const float*)d_in[0];
  const float* x      = (const float*)d_in[1];
  const float* mu     = (const float*)d_in[2];
  const float* logvar = (const float*)d_in[3];
  float* out = (float*)d_out;

  unsigned* minx_u = (unsigned*)d_ws;            // 32768 u32
  unsigned* miny_u = minx_u + NBATCH * NPTS;     // 32768 u32

  aether_init_kernel<<<(2 * NBATCH * NPTS) / 256, 256, 0, stream>>>(minx_u);
  aether_chamfer_kernel<<<NBATCH * ROWCS * COLCS, 256, 0, stream>>>(recon, x, minx_u, miny_u);
  aether_finalize_kernel<<<1, 256, 0, stream>>>(mu, logvar, minx_u, miny_u, out);
}